// TGCN_6966436954286
// MI455X (gfx1250) — compile-verified
//
#include <hip/hip_runtime.h>
#include <math.h>

#define B_  8
#define S_  12
#define N_  5000
#define F_  16
#define E_  40000
#define H_  128
#define L_  2
#define C_  4
#define TOTAL_ (B_*N_)          /* 40000 */
#define TH_    ((size_t)TOTAL_*H_)
#define BN_EPS_ 1e-5f

typedef __attribute__((ext_vector_type(2))) float v2f;
typedef __attribute__((ext_vector_type(8))) float v8f;
typedef int tdm4i __attribute__((vector_size(16)));   // matches builtin param type

#if __has_builtin(__builtin_amdgcn_global_load_async_to_lds_b128) && \
    __has_builtin(__builtin_amdgcn_s_wait_asynccnt)
#define USE_ASYNC_LDS 1
#else
#define USE_ASYNC_LDS 0
#endif

#define AS_STRIDE 260   /* 1040B rows: 16B-aligned for B128 async, conflict-free */

// ---------------------------------------------------------------------------
// Graph normalization (base graph, reused for all B copies)
// ---------------------------------------------------------------------------
__global__ __launch_bounds__(256) void k_deg(const int* __restrict__ ei,
                                             const float* __restrict__ ew,
                                             float* __restrict__ deg) {
  int e = blockIdx.x * blockDim.x + threadIdx.x;
  if (e >= E_) return;
  unsafeAtomicAdd(&deg[ei[E_ + e]], ew[e]);   // segment-sum over col
}

__global__ __launch_bounds__(256) void k_dis(float* __restrict__ deg) {
  int n = blockIdx.x * blockDim.x + threadIdx.x;
  if (n >= N_) return;
  deg[n] = rsqrtf(deg[n] + 1.0f);             // +1 = self-loop weight
}

__global__ __launch_bounds__(256) void k_norm(const int* __restrict__ ei,
                                              const float* __restrict__ ew,
                                              const float* __restrict__ dis,
                                              float* __restrict__ nrm) {
  int e = blockIdx.x * blockDim.x + threadIdx.x;
  if (e >= E_ + N_) return;
  if (e < E_) {
    nrm[e] = dis[ei[e]] * ew[e] * dis[ei[E_ + e]];
  } else {
    float d = dis[e - E_];
    nrm[e] = d * d;                            // self-loop norm
  }
}

// ---------------------------------------------------------------------------
// CSR build over destination (col) of the base graph: count -> scan -> fill
// ---------------------------------------------------------------------------
__global__ __launch_bounds__(256) void k_count(const int* __restrict__ ei,
                                               int* __restrict__ cnt) {
  int e = blockIdx.x * blockDim.x + threadIdx.x;
  if (e >= E_) return;
  atomicAdd(&cnt[ei[E_ + e]], 1);
}

// single-block exclusive scan over N_ counts (wave shfl + cross-wave LDS)
__global__ __launch_bounds__(256) void k_scan(const int* __restrict__ cnt,
                                              int* __restrict__ off,
                                              int* __restrict__ cur) {
  __shared__ int carry;
  __shared__ int wsum[8];
  if (threadIdx.x == 0) carry = 0;
  __syncthreads();
  const int lane = threadIdx.x & 31;
  const int wv   = threadIdx.x >> 5;
  for (int base = 0; base < N_; base += 256) {
    int i = base + threadIdx.x;
    int v = (i < N_) ? cnt[i] : 0;
    int x = v;
#pragma unroll
    for (int o = 1; o < 32; o <<= 1) {
      int y = __shfl_up(x, o, 32);
      if (lane >= o) x += y;
    }
    if (lane == 31) wsum[wv] = x;
    __syncthreads();
    int add = carry;
    for (int w2 = 0; w2 < wv; ++w2) add += wsum[w2];
    int excl = x - v + add;
    if (i < N_) { off[i] = excl; cur[i] = excl; }
    __syncthreads();
    if (threadIdx.x == 255) {
      int tot = 0;
      for (int w2 = 0; w2 < 8; ++w2) tot += wsum[w2];
      carry += tot;
    }
    __syncthreads();
  }
  if (threadIdx.x == 0) off[N_] = E_;
}

__global__ __launch_bounds__(256) void k_fill(const int* __restrict__ ei,
                                              int* __restrict__ cur,
                                              int* __restrict__ elist) {
  int e = blockIdx.x * blockDim.x + threadIdx.x;
  if (e >= E_) return;
  int pos = atomicAdd(&cur[ei[E_ + e]], 1);
  elist[pos] = e;
}

// ---------------------------------------------------------------------------
// Input projection: xt = relu(x_seq[:,t] @ Win + bin), [TOTAL,128]
// ---------------------------------------------------------------------------
__global__ __launch_bounds__(256) void k_lin_in(const float* __restrict__ xs,
                                                const float* __restrict__ W,
                                                const float* __restrict__ bias,
                                                float* __restrict__ xt, int t) {
  size_t tid = (size_t)blockIdx.x * blockDim.x + threadIdx.x;
  int i  = (int)(tid >> 7);
  int ch = (int)(tid & 127);
  int b = i / N_;
  int n = i - b * N_;
  const float* xr = xs + (((size_t)b * S_ + t) * N_ + n) * F_;
  float acc = bias[ch];
#pragma unroll
  for (int f = 0; f < F_; ++f) acc += xr[f] * W[f * H_ + ch];
  xt[(size_t)i * H_ + ch] = fmaxf(acc, 0.0f);
}

// ---------------------------------------------------------------------------
// WMMA f32 GEMM: C[TOTAL,128] = [A1 | A2] (K=256) @ Bm[256,128]
// One block = 16 rows; 8 waves, each owns one 16-wide N tile.
// A tile staged in LDS with GLOBAL_LOAD_ASYNC_TO_LDS_B128 when available.
// ---------------------------------------------------------------------------
__global__ __launch_bounds__(256) void k_gemm_wmma(const float* __restrict__ A1,
                                                   const float* __restrict__ A2,
                                                   const float* __restrict__ Bm,
                                                   float* __restrict__ Cm) {
  __shared__ float As[16 * AS_STRIDE];
  const int m0  = blockIdx.x * 16;
  const int tid = threadIdx.x;

#if USE_ASYNC_LDS
  // 1024 x B128 transfers: rows m0..m0+15, K 0..255 (A1 | A2 virtual concat)
  for (int g = tid; g < 1024; g += 256) {
    int r = g >> 6;              // 0..15
    int c = (g & 63) * 4;        // 0..252, 16B-aligned groups
    const float* src = (c < 128) ? A1 + (size_t)(m0 + r) * H_ + c
                                 : A2 + (size_t)(m0 + r) * H_ + (c - 128);
    __builtin_amdgcn_global_load_async_to_lds_b128(
        (__attribute__((address_space(1))) tdm4i*)(uintptr_t)src,
        (__attribute__((address_space(3))) tdm4i*)(uintptr_t)(uint32_t)(uintptr_t)
            (As + r * AS_STRIDE + c),
        0, 0);
  }
  __builtin_amdgcn_s_wait_asynccnt(0);
  __syncthreads();
#else
  for (int f = tid; f < 16 * 256; f += 256) {
    int r = f >> 8, c = f & 255;
    float v = (c < 128) ? A1[(size_t)(m0 + r) * H_ + c]
                        : A2[(size_t)(m0 + r) * H_ + (c - 128)];
    As[r * AS_STRIDE + c] = v;
  }
  __syncthreads();
#endif

  const int wave = tid >> 5;
  const int lane = tid & 31;
  const int half = lane >> 4;                  // lane group (ISA fragment layout)
  const int ln   = lane & 15;
  const int n0   = wave * 16;

  v8f acc = {};
  for (int k = 0; k < 256; k += 4) {
    v2f a, b;
    // A 16x4 f32 fragment: lane L -> row L%16; VGPR0=K(2*half), VGPR1=K(2*half+1)
    a.x = As[ln * AS_STRIDE + k + 2 * half];
    a.y = As[ln * AS_STRIDE + k + 2 * half + 1];
    // B 4x16 fragment: row-striped across lanes, rows v and v+2 per VGPR
    b.x = Bm[(size_t)(k + 2 * half)     * H_ + n0 + ln];
    b.y = Bm[(size_t)(k + 2 * half + 1) * H_ + n0 + ln];
    acc = __builtin_amdgcn_wmma_f32_16x16x4_f32(false, a, false, b,
                                                (short)0, acc, false, false);
  }
#pragma unroll
  for (int v = 0; v < 8; ++v)
    Cm[(size_t)(m0 + v + 8 * half) * H_ + n0 + ln] = acc[v];
}

// ---------------------------------------------------------------------------
// Atomic-free CSR aggregation, fused with gate activations.
// One wave32 per batched node; lane owns 4 channels (float4).
// ---------------------------------------------------------------------------
__device__ __forceinline__ float4 agg_gather(const int* __restrict__ ei,
                                             const float* __restrict__ nrm,
                                             const int* __restrict__ off,
                                             const int* __restrict__ elist,
                                             const float* __restrict__ xw,
                                             int i, int lane) {
  int b = i / N_;
  int n = i - b * N_;
  int base = b * N_;
  float w0 = nrm[E_ + n];                      // self loop
  float4 v = reinterpret_cast<const float4*>(xw + (size_t)i * H_)[lane];
  float4 acc;
  acc.x = w0 * v.x; acc.y = w0 * v.y; acc.z = w0 * v.z; acc.w = w0 * v.w;
  int k1 = off[n + 1];
  for (int k = off[n]; k < k1; ++k) {
    int e   = elist[k];
    int row = ei[e] + base;
    float w = nrm[e];
    float4 s = reinterpret_cast<const float4*>(xw + (size_t)row * H_)[lane];
    acc.x += w * s.x; acc.y += w * s.y; acc.z += w * s.z; acc.w += w * s.w;
  }
  return acc;
}

__global__ __launch_bounds__(256) void k_agg_sigmoid(const int* __restrict__ ei,
                                                     const float* __restrict__ nrm,
                                                     const int* __restrict__ off,
                                                     const int* __restrict__ elist,
                                                     const float* __restrict__ xw,
                                                     const float* __restrict__ bias,
                                                     float* __restrict__ out) {
  size_t tid = (size_t)blockIdx.x * blockDim.x + threadIdx.x;
  int i = (int)(tid >> 5), lane = (int)(tid & 31);
  if (i >= TOTAL_) return;
  float4 acc = agg_gather(ei, nrm, off, elist, xw, i, lane);
  float4 bb  = reinterpret_cast<const float4*>(bias)[lane];
  float4 o;
  o.x = 1.0f / (1.0f + __expf(-(acc.x + bb.x)));
  o.y = 1.0f / (1.0f + __expf(-(acc.y + bb.y)));
  o.z = 1.0f / (1.0f + __expf(-(acc.z + bb.z)));
  o.w = 1.0f / (1.0f + __expf(-(acc.w + bb.w)));
  reinterpret_cast<float4*>(out + (size_t)i * H_)[lane] = o;
}

// candidate gate + GRU state update fused: h = u*h + (1-u)*tanh(agg + bias)
__global__ __launch_bounds__(256) void k_agg_update(const int* __restrict__ ei,
                                                    const float* __restrict__ nrm,
                                                    const int* __restrict__ off,
                                                    const int* __restrict__ elist,
                                                    const float* __restrict__ xw,
                                                    const float* __restrict__ bias,
                                                    const float* __restrict__ u,
                                                    float* __restrict__ h) {
  size_t tid = (size_t)blockIdx.x * blockDim.x + threadIdx.x;
  int i = (int)(tid >> 5), lane = (int)(tid & 31);
  if (i >= TOTAL_) return;
  float4 acc = agg_gather(ei, nrm, off, elist, xw, i, lane);
  float4 bb  = reinterpret_cast<const float4*>(bias)[lane];
  float4 uu  = reinterpret_cast<const float4*>(u + (size_t)i * H_)[lane];
  float4 hh  = reinterpret_cast<float4*>(h + (size_t)i * H_)[lane];
  hh.x = uu.x * hh.x + (1.0f - uu.x) * tanhf(acc.x + bb.x);
  hh.y = uu.y * hh.y + (1.0f - uu.y) * tanhf(acc.y + bb.y);
  hh.z = uu.z * hh.z + (1.0f - uu.z) * tanhf(acc.z + bb.z);
  hh.w = uu.w * hh.w + (1.0f - uu.w) * tanhf(acc.w + bb.w);
  reinterpret_cast<float4*>(h + (size_t)i * H_)[lane] = hh;
}

__global__ __launch_bounds__(256) void k_mul_inplace(float* __restrict__ r,
                                                     const float* __restrict__ h) {
  size_t tid = (size_t)blockIdx.x * blockDim.x + threadIdx.x;
  r[tid] = r[tid] * h[tid];
}

// ---------------------------------------------------------------------------
// BatchNorm statistics (sum / sumsq per channel) + fused head
// ---------------------------------------------------------------------------
__global__ __launch_bounds__(128) void k_bn_stats(const float* __restrict__ h,
                                                  float* __restrict__ sums) {
  int ch = threadIdx.x;
  int r0 = blockIdx.x * 128;
  int r1 = r0 + 128; if (r1 > TOTAL_) r1 = TOTAL_;
  float s = 0.0f, s2 = 0.0f;
  for (int r = r0; r < r1; ++r) {
    float v = h[(size_t)r * H_ + ch];
    s += v; s2 += v * v;
  }
  unsafeAtomicAdd(&sums[ch], s);
  unsafeAtomicAdd(&sums[128 + ch], s2);
}

// BN -> relu -> [128,4] linear -> log_softmax ; one wave32 per node
__global__ __launch_bounds__(256) void k_head(const float* __restrict__ h,
                                              const float* __restrict__ sums,
                                              const float* __restrict__ gamma,
                                              const float* __restrict__ beta,
                                              const float* __restrict__ Wout,
                                              const float* __restrict__ bout,
                                              float* __restrict__ out) {
  size_t tid = (size_t)blockIdx.x * blockDim.x + threadIdx.x;
  int node = (int)(tid >> 5);
  int lane = (int)(tid & 31);
  if (node >= TOTAL_) return;
  const float inv = 1.0f / (float)TOTAL_;
  float a0 = 0.f, a1 = 0.f, a2 = 0.f, a3 = 0.f;
#pragma unroll
  for (int j = 0; j < 4; ++j) {
    int ch = lane + 32 * j;
    float mean = sums[ch] * inv;
    float var  = sums[128 + ch] * inv - mean * mean;
    float v = (h[(size_t)node * H_ + ch] - mean) * rsqrtf(var + BN_EPS_)
              * gamma[ch] + beta[ch];
    v = fmaxf(v, 0.0f);
    a0 += v * Wout[ch * 4 + 0];
    a1 += v * Wout[ch * 4 + 1];
    a2 += v * Wout[ch * 4 + 2];
    a3 += v * Wout[ch * 4 + 3];
  }
#pragma unroll
  for (int off = 16; off >= 1; off >>= 1) {
    a0 += __shfl_xor(a0, off, 32);
    a1 += __shfl_xor(a1, off, 32);
    a2 += __shfl_xor(a2, off, 32);
    a3 += __shfl_xor(a3, off, 32);
  }
  if (lane == 0) {
    float l0 = a0 + bout[0], l1 = a1 + bout[1], l2 = a2 + bout[2], l3 = a3 + bout[3];
    float m  = fmaxf(fmaxf(l0, l1), fmaxf(l2, l3));
    float s  = __expf(l0 - m) + __expf(l1 - m) + __expf(l2 - m) + __expf(l3 - m);
    float lse = m + __logf(s);
    float* o = out + (size_t)node * C_;
    o[0] = l0 - lse; o[1] = l1 - lse; o[2] = l2 - lse; o[3] = l3 - lse;
  }
}

// ---------------------------------------------------------------------------
// Host orchestration
// ---------------------------------------------------------------------------
extern "C" void kernel_launch(void* const* d_in, const int* in_sizes, int n_in,
                              void* d_out, int out_size, void* d_ws, size_t ws_size,
                              hipStream_t stream) {
  (void)in_sizes; (void)n_in; (void)out_size; (void)ws_size;
  const float* x_seq  = (const float*)d_in[0];
  const int*   ei     = (const int*)  d_in[1];   // [2,E]: rows then cols
  const float* ew     = (const float*)d_in[2];
  const float* Win    = (const float*)d_in[3];
  const float* bin    = (const float*)d_in[4];
  const float* convW  = (const float*)d_in[5];   // [L,3,256,128]
  const float* convB  = (const float*)d_in[6];   // [L,3,128]
  const float* gamma  = (const float*)d_in[7];
  const float* beta   = (const float*)d_in[8];
  const float* Wout   = (const float*)d_in[9];
  const float* bout   = (const float*)d_in[10];
  float*       out    = (float*)d_out;

  float* ws    = (float*)d_ws;
  float* deg   = ws;                    // 5120
  float* nrm   = deg + 5120;            // 46080 (E+N used)
  float* bnsum = nrm + 46080;           // 1024 (256 used)
  int*   cnt   = (int*)(bnsum + 1024);  // 5120
  int*   off   = cnt + 5120;            // 5120 (N+1 used)
  int*   cur   = off + 5120;            // 5120
  int*   elist = cur + 5120;            // 40960 (E used)
  float* xt    = (float*)(elist + 40960);
  float* h0    = xt  + TH_;
  float* h1    = h0  + TH_;
  float* xw    = h1  + TH_;
  float* rbuf  = xw  + TH_;
  float* ubuf  = rbuf + TH_;

  const int thr = 256;
  const int gElem = (int)(TH_ / thr);                      // 20000
  const int gGemm = TOTAL_ / 16;                           // 2500
  const int gNode = (int)(((size_t)TOTAL_ * 32) / thr);    // 5000 (wave per node)

  // --- graph norm + CSR (base graph) ---
  (void)hipMemsetAsync(deg, 0, 5120 * sizeof(float), stream);
  (void)hipMemsetAsync(cnt, 0, 5120 * sizeof(int), stream);
  k_deg  <<<(E_ + thr - 1) / thr, thr, 0, stream>>>(ei, ew, deg);
  k_dis  <<<(N_ + thr - 1) / thr, thr, 0, stream>>>(deg);
  k_norm <<<(E_ + N_ + thr - 1) / thr, thr, 0, stream>>>(ei, ew, deg, nrm);
  k_count<<<(E_ + thr - 1) / thr, thr, 0, stream>>>(ei, cnt);
  k_scan <<<1, thr, 0, stream>>>(cnt, off, cur);
  k_fill <<<(E_ + thr - 1) / thr, thr, 0, stream>>>(ei, cur, elist);

  // --- zero hidden state (every call: determinism) ---
  (void)hipMemsetAsync(h0, 0, TH_ * sizeof(float), stream);
  (void)hipMemsetAsync(h1, 0, TH_ * sizeof(float), stream);

  for (int t = 0; t < S_; ++t) {
    k_lin_in<<<gElem, thr, 0, stream>>>(x_seq, Win, bin, xt, t);
    for (int l = 0; l < L_; ++l) {
      const float* xin = (l == 0) ? xt : h0;
      float*       hl  = (l == 0) ? h0 : h1;
      const float* W0 = convW + ((size_t)(l * 3 + 0)) * (2 * H_ * H_);
      const float* W1 = convW + ((size_t)(l * 3 + 1)) * (2 * H_ * H_);
      const float* W2 = convW + ((size_t)(l * 3 + 2)) * (2 * H_ * H_);
      const float* B0 = convB + (l * 3 + 0) * H_;
      const float* B1 = convB + (l * 3 + 1) * H_;
      const float* B2 = convB + (l * 3 + 2) * H_;

      // r gate
      k_gemm_wmma <<<gGemm, thr, 0, stream>>>(xin, hl, W0, xw);
      k_agg_sigmoid<<<gNode, thr, 0, stream>>>(ei, nrm, off, elist, xw, B0, rbuf);
      // u gate
      k_gemm_wmma <<<gGemm, thr, 0, stream>>>(xin, hl, W1, xw);
      k_agg_sigmoid<<<gNode, thr, 0, stream>>>(ei, nrm, off, elist, xw, B1, ubuf);
      // candidate uses [x | r*h]; then fused h = u*h + (1-u)*tanh(...)
      k_mul_inplace<<<gElem, thr, 0, stream>>>(rbuf, hl);
      k_gemm_wmma <<<gGemm, thr, 0, stream>>>(xin, rbuf, W2, xw);
      k_agg_update<<<gNode, thr, 0, stream>>>(ei, nrm, off, elist, xw, B2, ubuf, hl);
    }
  }

  // --- BN (training-mode stats) + relu + linear head + log_softmax ---
  (void)hipMemsetAsync(bnsum, 0, 1024 * sizeof(float), stream);
  k_bn_stats<<<(TOTAL_ + 127) / 128, 128, 0, stream>>>(h1, bnsum);
  k_head<<<gNode, thr, 0, stream>>>(h1, bnsum, gamma, beta, Wout, bout, out);
}